// GNNSensorsAndConditions_74921409511965
// MI455X (gfx1250) — compile-verified
//
#include <hip/hip_runtime.h>
#include <hip/hip_bf16.h>

typedef __attribute__((ext_vector_type(2))) float v2f;
typedef __attribute__((ext_vector_type(8))) float v8f;

#define NPG 18   // nodes per graph
#define NSEN 14  // sensor nodes per graph

// ---------------------------------------------------------------------------
// deg init: deg1[0..N) and deg2[0..Ns) are contiguous -> fill N+Ns with 1.0f
// ---------------------------------------------------------------------------
__global__ void k_fill1(float* __restrict__ p, int n) {
    int i = blockIdx.x * blockDim.x + threadIdx.x;
    if (i < n) p[i] = 1.0f;
}

// one thread per edge: accumulate weighted degree for conv1 (all edges) and
// conv2 (sensor-sensor edges, remapped index space)
__global__ void k_deg(const int* __restrict__ src, const int* __restrict__ dst,
                      const float* __restrict__ ew,
                      float* __restrict__ deg1, float* __restrict__ deg2, int E) {
    int e = blockIdx.x * blockDim.x + threadIdx.x;
    if (e >= E) return;
    int s = src[e], d = dst[e];
    float w = ew[e];
    atomicAdd(&deg1[d], w);
    int sn = s % NPG, dn = d % NPG;
    if (sn < NSEN && dn < NSEN) {
        int nd = (d / NPG) * NSEN + dn;
        atomicAdd(&deg2[nd], w);
    }
}

// in-place deg -> 1/sqrt(deg)  (deg >= 1 always due to self-loop weight 1)
__global__ void k_rsqrt(float* __restrict__ p, int n) {
    int i = blockIdx.x * blockDim.x + threadIdx.x;
    if (i < n) {
        float d = p[i];
        p[i] = (d > 0.0f) ? rsqrtf(d) : 0.0f;
    }
}

// ---------------------------------------------------------------------------
// GEMM1: H[N,128] = X[N,50] @ W1[50,128], fp32 WMMA 16x16x4.
// Block = 128 threads = 4 waves; each wave computes two 16x16 column tiles.
//
// WMMA fragment layouts (ISA 7.12.2), lane l, m=l&15, kh=l>>4:
//   A (16x4):  element v has (row m, K = v + 2*kh)     -> contiguous K pair
//   B (4x16):  element v has (K = v + 2*kh, col m)     -> K pair across rows
//   C (16x16): element v is (row v + 8*kh, col m)
//
// B is stored K-pair interleaved in LDS: Wl[(k>>1)*PSTRIDE + n*2 + (k&1)],
// so each lane's (ka, ka+1) pair is one aligned ds_load_b64 straight into an
// even VGPR pair (no repack movs). PSTRIDE padded so lane halves (pair-row
// and pair-row+1) hit disjoint banks. K zero-padded 50 -> 52: branch-free loop.
// ---------------------------------------------------------------------------
#define PS1 (128 * 2 + 2)  // 258: pair-row stride for W1 (26 pairs)
__global__ __launch_bounds__(128) void k_gemm1(const float* __restrict__ X,
                                               const float* __restrict__ W1,
                                               float* __restrict__ H, int Nrows) {
    __shared__ float Wl[26 * PS1];  // K pairs 0..25 (K up to 52, zero padded)
    __shared__ float Xl[16 * 52];   // 16-row X tile, stride 52, cols 50/51 = 0

    const int tid = threadIdx.x;
    for (int idx = tid; idx < 52 * 128; idx += 128) {
        int k = idx >> 7, n = idx & 127;
        Wl[(k >> 1) * PS1 + n * 2 + (k & 1)] = (k < 50) ? W1[(size_t)k * 128 + n] : 0.0f;
    }
    const int m0 = blockIdx.x * 16;
    for (int idx = tid; idx < 16 * 52; idx += 128) {
        int r = idx / 52, c = idx - r * 52;
        int row = m0 + r; if (row >= Nrows) row = Nrows - 1;
        Xl[r * 52 + c] = (c < 50) ? X[(size_t)row * 50 + c] : 0.0f;
    }
    __syncthreads();

    const int w  = tid >> 5;       // wave id 0..3
    const int l  = tid & 31;
    const int m  = l & 15;
    const int kh = l >> 4;         // lane half
    const int n0a = w * 16;
    const int n0b = (w + 4) * 16;

    v8f acc0 = {}; v8f acc1 = {};
    for (int k0 = 0; k0 < 52; k0 += 4) {   // 13 branch-free iterations
        const int kp = (k0 >> 1) + kh;     // K-pair index for this lane half
        v2f a  = *(const v2f*)&Xl[m * 52 + 2 * kp];
        v2f b0 = *(const v2f*)&Wl[kp * PS1 + (n0a + m) * 2];
        v2f b1 = *(const v2f*)&Wl[kp * PS1 + (n0b + m) * 2];
        acc0 = __builtin_amdgcn_wmma_f32_16x16x4_f32(false, a, false, b0, (short)0, acc0, false, false);
        acc1 = __builtin_amdgcn_wmma_f32_16x16x4_f32(false, a, false, b1, (short)0, acc1, false, false);
    }
#pragma unroll
    for (int v = 0; v < 8; ++v) {
        int row = m0 + v + 8 * kh;
        if (row < Nrows) {
            H[(size_t)row * 128 + n0a + m] = acc0[v];
            H[(size_t)row * 128 + n0b + m] = acc1[v];
        }
    }
}

// ---------------------------------------------------------------------------
// conv1 scatter: 32 lanes per edge, each lane handles 4 of 128 features
// ---------------------------------------------------------------------------
__global__ void k_scatter1(const int* __restrict__ src, const int* __restrict__ dst,
                           const float* __restrict__ ew, const float* __restrict__ dinv1,
                           const float* __restrict__ H, float* __restrict__ out1, int E) {
    size_t t = (size_t)blockIdx.x * blockDim.x + threadIdx.x;
    int e = (int)(t >> 5);
    int j = (int)(t & 31);
    if (e >= E) return;
    int s = src[e], d = dst[e];
    float norm = dinv1[s] * ew[e] * dinv1[d];
    const float* hs = H + (size_t)s * 128;
    float* od = out1 + (size_t)d * 128;
#pragma unroll
    for (int c = 0; c < 4; ++c)
        atomicAdd(&od[j + 32 * c], hs[j + 32 * c] * norm);
}

// h1 = relu(scatter + self_loop + bias), in place in out1
__global__ void k_fin1(float* __restrict__ out1, const float* __restrict__ H,
                       const float* __restrict__ dinv1, const float* __restrict__ b1,
                       size_t total) {
    size_t i = (size_t)blockIdx.x * blockDim.x + threadIdx.x;
    if (i >= total) return;
    size_t row = i >> 7;
    int f = (int)(i & 127);
    float di = dinv1[row];
    float v = out1[i] + H[i] * di * di + b1[f];
    out1[i] = v > 0.0f ? v : 0.0f;
}

// ---------------------------------------------------------------------------
// GEMM2: G[Ns,64] = HS[Ns,128] @ W2[128,64]; sensor-row gather fused into the
// LDS tile load. 4 waves, one 16x16 column tile each. K = 128 = 32 clean steps.
// Same K-pair-interleaved B layout as GEMM1.
// ---------------------------------------------------------------------------
#define PS2 (64 * 2 + 2)   // 130: pair-row stride for W2 (64 pairs)
__global__ __launch_bounds__(128) void k_gemm2(const float* __restrict__ H1,
                                               const float* __restrict__ W2,
                                               float* __restrict__ G, int NsRows) {
    __shared__ float Wl[64 * PS2];  // 33.3 KB
    __shared__ float Hl[16 * 132];  // gathered HS tile, stride 132

    const int tid = threadIdx.x;
    for (int idx = tid; idx < 128 * 64; idx += 128) {
        int k = idx >> 6, n = idx & 63;
        Wl[(k >> 1) * PS2 + n * 2 + (k & 1)] = W2[idx];
    }
    const int m0 = blockIdx.x * 16;
    for (int r = 0; r < 16; ++r) {
        int i = m0 + r; if (i >= NsRows) i = NsRows - 1;
        int grow = (i / NSEN) * NPG + (i % NSEN);   // sensor -> full-graph row
        Hl[r * 132 + tid] = H1[(size_t)grow * 128 + tid];
    }
    __syncthreads();

    const int w  = tid >> 5;
    const int l  = tid & 31;
    const int m  = l & 15;
    const int kh = l >> 4;
    const int n0 = w * 16;

    v8f acc = {};
    for (int k0 = 0; k0 < 128; k0 += 4) {
        const int kp = (k0 >> 1) + kh;
        v2f a = *(const v2f*)&Hl[m * 132 + 2 * kp];
        v2f b = *(const v2f*)&Wl[kp * PS2 + (n0 + m) * 2];
        acc = __builtin_amdgcn_wmma_f32_16x16x4_f32(false, a, false, b, (short)0, acc, false, false);
    }
#pragma unroll
    for (int v = 0; v < 8; ++v) {
        int row = m0 + v + 8 * kh;
        if (row < NsRows) G[(size_t)row * 64 + n0 + m] = acc[v];
    }
}

// conv2 scatter: 16 lanes per edge, each lane handles 4 of 64 features
__global__ void k_scatter2(const int* __restrict__ src, const int* __restrict__ dst,
                           const float* __restrict__ ew, const float* __restrict__ dinv2,
                           const float* __restrict__ G, float* __restrict__ out2, int E) {
    size_t t = (size_t)blockIdx.x * blockDim.x + threadIdx.x;
    int e = (int)(t >> 4);
    int j = (int)(t & 15) * 4;
    if (e >= E) return;
    int s = src[e], d = dst[e];
    int sn = s % NPG, dn = d % NPG;
    if (sn >= NSEN || dn >= NSEN) return;   // dropped edges have weight 0 anyway
    int ns = (s / NPG) * NSEN + sn;
    int nd = (d / NPG) * NSEN + dn;
    float norm = dinv2[ns] * ew[e] * dinv2[nd];
    const float* gs = G + (size_t)ns * 64 + j;
    float* od = out2 + (size_t)nd * 64 + j;
    atomicAdd(&od[0], gs[0] * norm);
    atomicAdd(&od[1], gs[1] * norm);
    atomicAdd(&od[2], gs[2] * norm);
    atomicAdd(&od[3], gs[3] * norm);
}

__global__ void k_fin2(float* __restrict__ out2, const float* __restrict__ G,
                       const float* __restrict__ dinv2, const float* __restrict__ b2,
                       size_t total) {
    size_t i = (size_t)blockIdx.x * blockDim.x + threadIdx.x;
    if (i >= total) return;
    size_t row = i >> 6;
    int f = (int)(i & 63);
    float di = dinv2[row];
    float v = out2[i] + G[i] * di * di + b2[f];
    out2[i] = v > 0.0f ? v : 0.0f;
}

// per-node head: out[i] = dot(h2[i,:], Wh[i%14,:]) + bh[i%14]
__global__ void k_head(const float* __restrict__ h2, const float* __restrict__ Wh,
                       const float* __restrict__ bh, float* __restrict__ out, int Ns) {
    int i = blockIdx.x * blockDim.x + threadIdx.x;
    if (i >= Ns) return;
    int n = i % NSEN;
    const float4* a = (const float4*)(h2 + (size_t)i * 64);
    const float4* w = (const float4*)(Wh + (size_t)n * 64);
    float acc = bh[n];
#pragma unroll
    for (int k = 0; k < 16; ++k) {
        float4 xa = a[k], wa = w[k];
        acc += xa.x * wa.x + xa.y * wa.y + xa.z * wa.z + xa.w * wa.w;
    }
    out[i] = acc;
}

// ---------------------------------------------------------------------------
extern "C" void kernel_launch(void* const* d_in, const int* in_sizes, int n_in,
                              void* d_out, int out_size, void* d_ws, size_t ws_size,
                              hipStream_t stream) {
    const float* x  = (const float*)d_in[0];
    const int*   ei = (const int*)d_in[1];
    const float* ew = (const float*)d_in[2];
    const float* W1 = (const float*)d_in[3];
    const float* b1 = (const float*)d_in[4];
    const float* W2 = (const float*)d_in[5];
    const float* b2 = (const float*)d_in[6];
    const float* Wh = (const float*)d_in[7];
    const float* bh = (const float*)d_in[8];

    const int N  = in_sizes[0] / 50;        // total nodes
    const int E  = in_sizes[2];             // edges
    const int Ns = (N / NPG) * NSEN;        // sensor nodes
    const int* srcp = ei;
    const int* dstp = ei + E;

    // workspace layout (floats)
    float* h    = (float*)d_ws;               // [N,128]
    float* out1 = h    + (size_t)N * 128;     // [N,128]  (becomes h1)
    float* g    = out1 + (size_t)N * 128;     // [Ns,64]
    float* out2 = g    + (size_t)Ns * 64;     // [Ns,64]  (becomes h2)
    float* deg1 = out2 + (size_t)Ns * 64;     // [N]   (becomes dinv1)
    float* deg2 = deg1 + (size_t)N;           // [Ns]  (becomes dinv2)

    // zero accumulators (capture-legal stream memset)
    hipMemsetAsync(out1, 0, (size_t)N  * 128 * sizeof(float), stream);
    hipMemsetAsync(out2, 0, (size_t)Ns * 64  * sizeof(float), stream);

    const int T = 256;
    // degrees (deg1|deg2 contiguous)
    k_fill1<<<(N + Ns + T - 1) / T, T, 0, stream>>>(deg1, N + Ns);
    k_deg<<<(E + T - 1) / T, T, 0, stream>>>(srcp, dstp, ew, deg1, deg2, E);
    k_rsqrt<<<(N + Ns + T - 1) / T, T, 0, stream>>>(deg1, N + Ns);

    // conv1
    k_gemm1<<<(N + 15) / 16, 128, 0, stream>>>(x, W1, h, N);
    {
        size_t threads = (size_t)E * 32;
        k_scatter1<<<(unsigned)((threads + T - 1) / T), T, 0, stream>>>(
            srcp, dstp, ew, deg1, h, out1, E);
    }
    {
        size_t total = (size_t)N * 128;
        k_fin1<<<(unsigned)((total + T - 1) / T), T, 0, stream>>>(out1, h, deg1, b1, total);
    }

    // conv2 (out1 now holds h1)
    k_gemm2<<<(Ns + 15) / 16, 128, 0, stream>>>(out1, W2, g, Ns);
    {
        size_t threads = (size_t)E * 16;
        k_scatter2<<<(unsigned)((threads + T - 1) / T), T, 0, stream>>>(
            srcp, dstp, ew, deg2, g, out2, E);
    }
    {
        size_t total = (size_t)Ns * 64;
        k_fin2<<<(unsigned)((total + T - 1) / T), T, 0, stream>>>(out2, g, deg2, b2, total);
    }

    // head (out2 now holds h2)
    k_head<<<(Ns + T - 1) / T, T, 0, stream>>>(out2, Wh, bh, (float*)d_out, Ns);
}